// ModelEstimator_2783138808084
// MI455X (gfx1250) — compile-verified
//
#include <hip/hip_runtime.h>

// ---------------------------------------------------------------------------
// Weighted 8-point fundamental matrix estimation for MI455X (gfx1250).
// Pass 1: weighted centroids. Pass 2 (fused): weighted mean radius + 9x9 Gram
// matrix Sum(w^2 q q^T) accumulated with V_WMMA_F32_16X16X4_F32 (f32 WMMA).
// Pass 3: per-batch Jacobi eigensolves + rank-2 projection + denormalization.
// ---------------------------------------------------------------------------

typedef __attribute__((ext_vector_type(2))) float v2f;
typedef __attribute__((ext_vector_type(8))) float v8f;

#define AUG 1e-4f
#define BATCH 64
#define NPTS 32768
// workspace layout (floats): per batch 8 stats [sw,sx1,sy1,sx2,sy2,swr1,swr2,pad]
// then per batch 81 floats of the (unscaled) Gram matrix Mq.
#define WS_STATS(b) ((b) * 8)
#define WS_M(b) (BATCH * 8 + (b) * 81)

// ---------------------------------------------------------------------------
// Kernel 1: per-batch weighted sums for the centroids.
// ---------------------------------------------------------------------------
__global__ __launch_bounds__(256) void fe_stats_kernel(
    const float* __restrict__ pts1, const float* __restrict__ pts2,
    const float* __restrict__ wts, float* __restrict__ ws) {
  __shared__ float s[5][256];
  const int b = blockIdx.x;
  const int tid = threadIdx.x;
  const float* p1 = pts1 + (size_t)b * NPTS * 3;
  const float* p2 = pts2 + (size_t)b * NPTS * 3;
  const float* w = wts + (size_t)b * NPTS;

  float sw = 0.f, sx1 = 0.f, sy1 = 0.f, sx2 = 0.f, sy2 = 0.f;
  for (int n = tid; n < NPTS; n += 256) {
    float wn = w[n];
    float x1 = p1[n * 3 + 0], y1 = p1[n * 3 + 1];
    float x2 = p2[n * 3 + 0], y2 = p2[n * 3 + 1];
    sw += wn;
    sx1 += wn * x1;
    sy1 += wn * y1;
    sx2 += wn * x2;
    sy2 += wn * y2;
  }
  s[0][tid] = sw; s[1][tid] = sx1; s[2][tid] = sy1; s[3][tid] = sx2; s[4][tid] = sy2;
  __syncthreads();
  for (int off = 128; off > 0; off >>= 1) {
    if (tid < off) {
#pragma unroll
      for (int k = 0; k < 5; ++k) s[k][tid] += s[k][tid + off];
    }
    __syncthreads();
  }
  if (tid == 0) {
#pragma unroll
    for (int k = 0; k < 5; ++k) ws[WS_STATS(b) + k] = s[k][0];
  }
}

// ---------------------------------------------------------------------------
// Kernel 2: fused pass — weighted mean radius + WMMA Gram accumulation.
// One block (8 waves) per batch; each wave processes 32 points/iteration and
// feeds them through 8 x V_WMMA_F32_16X16X4_F32 (K = 4 points each).
// ---------------------------------------------------------------------------
__global__ __launch_bounds__(256) void fe_accum_kernel(
    const float* __restrict__ pts1, const float* __restrict__ pts2,
    const float* __restrict__ wts, float* __restrict__ ws) {
  // q components zero-padded to 16 so fragment gathers are branch-free.
  __shared__ __align__(16) float sQ[8][32][16];   // q per point
  __shared__ __align__(16) float sWQ[8][32][16];  // w^2 * q per point
  __shared__ float red[8][256];                   // cross-wave accum reduce
  __shared__ float swr[2][256];                   // mean-radius reduce

  const int b = blockIdx.x;
  const int tid = threadIdx.x;
  const int wave = tid >> 5;
  const int lane = tid & 31;
  const int comp = lane & 15;  // matrix row (A) / col (B) index
  const int hlf = lane >> 4;   // K-half selector per ISA 16x4 f32 layout

  const float* p1 = pts1 + (size_t)b * NPTS * 3;
  const float* p2 = pts2 + (size_t)b * NPTS * 3;
  const float* w = wts + (size_t)b * NPTS;

  const float sw = ws[WS_STATS(b) + 0];
  const float cx1 = ws[WS_STATS(b) + 1] / sw;
  const float cy1 = ws[WS_STATS(b) + 2] / sw;
  const float cx2 = ws[WS_STATS(b) + 3] / sw;
  const float cy2 = ws[WS_STATS(b) + 4] / sw;

  float wr1 = 0.f, wr2 = 0.f;
  v8f acc = {0.f, 0.f, 0.f, 0.f, 0.f, 0.f, 0.f, 0.f};

  const int iters = NPTS / 256;  // 128
  for (int it = 0; it < iters; ++it) {
    const int n = it * 256 + tid;
    const float wn = w[n];
    const float u1 = p1[n * 3 + 0] - cx1;
    const float v1 = p1[n * 3 + 1] - cy1;
    const float u2 = p2[n * 3 + 0] - cx2;
    const float v2 = p2[n * 3 + 1] - cy2;

    if (it + 1 < iters) {  // gfx1250 global_prefetch_b8 of next tile
      __builtin_prefetch(&p1[(n + 256) * 3], 0, 1);
      __builtin_prefetch(&p2[(n + 256) * 3], 0, 1);
      __builtin_prefetch(&w[n + 256], 0, 1);
    }

    wr1 += wn * sqrtf(u1 * u1 + v1 * v1);
    wr2 += wn * sqrtf(u2 * u2 + v2 * v2);

    const float w2 = wn * wn;
    float q[16] = {u1 * u2, u1 * v2, u1, v1 * u2, v1 * v2, v1, u2, v2, 1.f,
                   0.f, 0.f, 0.f, 0.f, 0.f, 0.f, 0.f};
    float4* dq = (float4*)&sQ[wave][lane][0];
    float4* dwq = (float4*)&sWQ[wave][lane][0];
#pragma unroll
    for (int k = 0; k < 4; ++k) {
      dq[k] = make_float4(q[4 * k], q[4 * k + 1], q[4 * k + 2], q[4 * k + 3]);
      dwq[k] = make_float4(w2 * q[4 * k], w2 * q[4 * k + 1],
                           w2 * q[4 * k + 2], w2 * q[4 * k + 3]);
    }
    __syncthreads();

#pragma unroll
    for (int j = 0; j < 8; ++j) {
      const int pt = 4 * j + 2 * hlf;  // K = 2*half + vgpr (ISA 16x4 layout)
      v2f va, vb;
      va.x = sQ[wave][pt][comp];
      va.y = sQ[wave][pt + 1][comp];
      vb.x = sWQ[wave][pt][comp];
      vb.y = sWQ[wave][pt + 1][comp];
      // C(16x16) += A(16x4) * B(4x16): M[m][n] += sum_k q_k[m] * (w^2 q_k)[n]
      acc = __builtin_amdgcn_wmma_f32_16x16x4_f32(
          false, va, false, vb, (short)0, acc, false, false);
    }
    __syncthreads();
  }

  // reduce mean-radius sums across the block
  swr[0][tid] = wr1;
  swr[1][tid] = wr2;
  // stage per-wave WMMA accumulators
#pragma unroll
  for (int r = 0; r < 8; ++r) red[wave][lane * 8 + r] = acc[r];
  __syncthreads();
  for (int off = 128; off > 0; off >>= 1) {
    if (tid < off) {
      swr[0][tid] += swr[0][tid + off];
      swr[1][tid] += swr[1][tid + off];
    }
    __syncthreads();
  }
  if (tid == 0) {
    ws[WS_STATS(b) + 5] = swr[0][0];
    ws[WS_STATS(b) + 6] = swr[1][0];
  }
  // combine the 8 waves' C fragments and scatter valid 9x9 entries
  float sum = 0.f;
#pragma unroll
  for (int wv = 0; wv < 8; ++wv) sum += red[wv][tid];
  const int l = tid >> 3, r = tid & 7;
  const int row = r + 8 * (l >> 4);  // C layout: M = vgpr + 8*(lane>>4)
  const int col = l & 15;            // N = lane & 15
  if (row < 9 && col < 9) ws[WS_M(b) + row * 9 + col] = sum;
}

// ---------------------------------------------------------------------------
// Kernel 3: per-batch eigensolves + rank-2 projection + denormalization.
// ---------------------------------------------------------------------------
__device__ void jacobi_eigh(float* A, float* V, int n, int sweeps) {
  for (int i = 0; i < n; ++i)
    for (int j = 0; j < n; ++j) V[i * n + j] = (i == j) ? 1.f : 0.f;
  for (int s = 0; s < sweeps; ++s) {
    for (int p = 0; p < n - 1; ++p) {
      for (int q = p + 1; q < n; ++q) {
        float apq = A[p * n + q];
        if (fabsf(apq) < 1e-24f) continue;
        float theta = 0.5f * (A[q * n + q] - A[p * n + p]) / apq;
        float t = copysignf(1.f, theta) /
                  (fabsf(theta) + sqrtf(theta * theta + 1.f));
        float c = 1.f / sqrtf(t * t + 1.f);
        float sn = t * c;
        for (int k = 0; k < n; ++k) {  // A <- A J (columns p,q)
          float akp = A[k * n + p], akq = A[k * n + q];
          A[k * n + p] = c * akp - sn * akq;
          A[k * n + q] = sn * akp + c * akq;
        }
        for (int k = 0; k < n; ++k) {  // A <- J^T A (rows p,q)
          float apk = A[p * n + k], aqk = A[q * n + k];
          A[p * n + k] = c * apk - sn * aqk;
          A[q * n + k] = sn * apk + c * aqk;
        }
        for (int k = 0; k < n; ++k) {  // V <- V J
          float vkp = V[k * n + p], vkq = V[k * n + q];
          V[k * n + p] = c * vkp - sn * vkq;
          V[k * n + q] = sn * vkp + c * vkq;
        }
      }
    }
  }
}

__device__ void mat3mul(float* D, const float* A, const float* B) {
  for (int i = 0; i < 3; ++i)
    for (int j = 0; j < 3; ++j)
      D[i * 3 + j] = A[i * 3 + 0] * B[0 * 3 + j] + A[i * 3 + 1] * B[1 * 3 + j] +
                     A[i * 3 + 2] * B[2 * 3 + j];
}

__global__ __launch_bounds__(64) void fe_solve_kernel(
    const float* __restrict__ ws, float* __restrict__ out) {
  const int b = threadIdx.x;
  if (b >= BATCH) return;

  const float sw = ws[WS_STATS(b) + 0];
  const float cx1 = ws[WS_STATS(b) + 1] / sw;
  const float cy1 = ws[WS_STATS(b) + 2] / sw;
  const float cx2 = ws[WS_STATS(b) + 3] / sw;
  const float cy2 = ws[WS_STATS(b) + 4] / sw;
  const float s1 = 1.4142f / (ws[WS_STATS(b) + 5] / sw);
  const float s2 = 1.4142f / (ws[WS_STATS(b) + 6] / sw);

  // M = D Mq D + AUG I, with p = D q
  const float d[9] = {s1 * s2, s1 * s2, s1, s1 * s2, s1 * s2, s1, s2, s2, 1.f};
  float A[81], V[81];
  const float* Mq = ws + WS_M(b);
  for (int i = 0; i < 9; ++i)
    for (int j = 0; j < 9; ++j)
      A[i * 9 + j] = d[i] * d[j] * Mq[i * 9 + j] + ((i == j) ? AUG : 0.f);

  jacobi_eigh(A, V, 9, 15);
  int kmin = 0;
  float emin = A[0];
  for (int k = 1; k < 9; ++k)
    if (A[k * 9 + k] < emin) { emin = A[k * 9 + k]; kmin = k; }
  float F[9];
  for (int i = 0; i < 9; ++i) F[i] = V[i * 9 + kmin];  // nullspace -> 3x3

  // rank-2 projection: eigh(F F^T + AUG I); P = I - u u^T (u = smallest eigvec)
  float G[9], U[9];
  for (int i = 0; i < 3; ++i)
    for (int j = 0; j < 3; ++j) {
      float s = 0.f;
      for (int k = 0; k < 3; ++k) s += F[i * 3 + k] * F[j * 3 + k];
      G[i * 3 + j] = s + ((i == j) ? AUG : 0.f);
    }
  jacobi_eigh(G, U, 3, 10);
  int k3 = 0;
  float e3 = G[0];
  for (int k = 1; k < 3; ++k)
    if (G[k * 3 + k] < e3) { e3 = G[k * 3 + k]; k3 = k; }
  const float u[3] = {U[0 * 3 + k3], U[1 * 3 + k3], U[2 * 3 + k3]};
  float uf[3];
  for (int j = 0; j < 3; ++j)
    uf[j] = u[0] * F[0 * 3 + j] + u[1] * F[1 * 3 + j] + u[2] * F[2 * 3 + j];
  float Fp[9];
  for (int i = 0; i < 3; ++i)
    for (int j = 0; j < 3; ++j) Fp[i * 3 + j] = F[i * 3 + j] - u[i] * uf[j];

  // out = T1^T Fp T2
  const float T1t[9] = {s1, 0.f, 0.f, 0.f, s1, 0.f, -cx1 * s1, -cy1 * s1, 1.f};
  const float T2[9] = {s2, 0.f, -cx2 * s2, 0.f, s2, -cy2 * s2, 0.f, 0.f, 1.f};
  float H[9], O[9];
  mat3mul(H, T1t, Fp);
  mat3mul(O, H, T2);
  for (int i = 0; i < 9; ++i) out[b * 9 + i] = O[i];
}

// ---------------------------------------------------------------------------
extern "C" void kernel_launch(void* const* d_in, const int* in_sizes, int n_in,
                              void* d_out, int out_size, void* d_ws,
                              size_t ws_size, hipStream_t stream) {
  const float* pts1 = (const float*)d_in[0];  // (64, 32768, 3) f32
  const float* pts2 = (const float*)d_in[1];  // (64, 32768, 3) f32
  const float* wts = (const float*)d_in[2];   // (64, 1, 32768) f32
  float* out = (float*)d_out;                 // (64, 3, 3) f32
  float* ws = (float*)d_ws;

  fe_stats_kernel<<<BATCH, 256, 0, stream>>>(pts1, pts2, wts, ws);
  fe_accum_kernel<<<BATCH, 256, 0, stream>>>(pts1, pts2, wts, ws);
  fe_solve_kernel<<<1, 64, 0, stream>>>(ws, out);
}